// FireflyAlgorithm_50964081934653
// MI455X (gfx1250) — compile-verified
//
#include <hip/hip_runtime.h>
#include <hip/hip_bf16.h>
#include <math.h>

typedef float v2f __attribute__((ext_vector_type(2)));
typedef float v8f __attribute__((ext_vector_type(8)));

#define BATCH   4
#define SEQ     2048
#define HH      4096
#define NF      35
#define ROWS    (BATCH*SEQ)      // 8192
#define NT      3                // 16-wide column tiles -> 48 >= 35
#define NC      (NT*16)          // 48
#define KSTEPS  (HH/4)           // 1024 wmma K-steps (K=4 each)
#define KCHUNK  128
#define LDSS    132              // padded LDS row stride (banks: 132%64=4 -> conflict free)

// ---------------------------------------------------------------------------
// Kernel 0a: pack positions^T into WMMA-B fragment order.
// B[k][n] = P[n][k]; lane layout: lanes 0-15 hold N=lane, K={k,k+1};
// lanes 16-31 hold N=lane-16, K={k+2,k+3}. Stored lane-major per k-step so the
// hot-loop load is one fully-coalesced 256B float2 fetch per wave per step.
// ---------------------------------------------------------------------------
__global__ void ff_pack(const float* __restrict__ pos, float2* __restrict__ bpack) {
    int idx = blockIdx.x * blockDim.x + threadIdx.x;
    if (idx >= NT * KSTEPS * 32) return;
    int lane = idx & 31;
    int ks   = (idx >> 5) & (KSTEPS - 1);
    int t    = idx / (32 * KSTEPS);
    int n    = t * 16 + (lane & 15);
    int k    = ks * 4 + ((lane >> 4) << 1);
    float2 v = make_float2(0.f, 0.f);
    if (n < NF) {
        v.x = pos[(size_t)n * HH + k];
        v.y = pos[(size_t)n * HH + k + 1];
    }
    bpack[idx] = v;
}

// Kernel 0b: pnorm2[i] = ||p_i||^2
__global__ void ff_pnorm(const float* __restrict__ pos, float* __restrict__ pn2) {
    __shared__ float red[256];
    int i = blockIdx.x, tid = threadIdx.x;
    float s = 0.f;
    for (int k = tid; k < HH; k += 256) {
        float p = pos[(size_t)i * HH + k];
        s = fmaf(p, p, s);
    }
    red[tid] = s; __syncthreads();
    for (int o = 128; o > 0; o >>= 1) { if (tid < o) red[tid] += red[tid + o]; __syncthreads(); }
    if (tid == 0) pn2[i] = red[0];
}

// ---------------------------------------------------------------------------
// Kernel 1: dots[8192][48] = X * P^T via V_WMMA_F32_16X16X4_F32, plus row
// norms ||x_row||^2. 4 waves/block: waves 0-2 own one 16x16 N-tile each,
// wave 3 computes row norms. X tile (16 x 128) staged once in LDS per chunk.
// ---------------------------------------------------------------------------
__global__ __launch_bounds__(128)
void ff_gemm_norm(const float* __restrict__ x, const float2* __restrict__ bpack,
                  float* __restrict__ dots, float* __restrict__ xnorm2) {
    __shared__ float As[16 * LDSS];
    const int tile = blockIdx.x;
    const int tid  = threadIdx.x;
    const int wave = tid >> 5, lane = tid & 31;
    const int m  = lane & 15;
    const int kh = (lane >> 4) << 1;          // 0 or 2
    v8f  acc  = {};
    float nacc = 0.f;
    const float* xrow = x + (size_t)tile * 16 * HH;

    for (int kb = 0; kb < HH; kb += KCHUNK) {
        // cooperative load: 16 x 128 f32 chunk, float4, conflict-free store
        for (int u = tid; u < 16 * KCHUNK / 4; u += 128) {
            int r = u >> 5;
            int c = (u & 31) << 2;
            float4 v = *reinterpret_cast<const float4*>(xrow + (size_t)r * HH + kb + c);
            *reinterpret_cast<float4*>(&As[r * LDSS + c]) = v;
        }
        __syncthreads();
        if (wave < 3) {
            const float2* bp = bpack + ((size_t)wave * KSTEPS + (kb >> 2)) * 32 + lane;
            #pragma unroll
            for (int ks = 0; ks < KCHUNK / 4; ++ks) {
                int k = ks * 4;
                v2f a = *reinterpret_cast<const v2f*>(&As[m * LDSS + k + kh]);
                float2 braw = bp[ks * 32];
                v2f b = {braw.x, braw.y};
                acc = __builtin_amdgcn_wmma_f32_16x16x4_f32(
                        false, a, false, b, (short)0, acc, false, false);
            }
        } else {
            // row norms: lane pair (h=0/1) covers interleaved float2s of row m
            #pragma unroll
            for (int t = 0; t < 32; ++t) {
                int k = kh + t * 4;
                float2 v = *reinterpret_cast<const float2*>(&As[m * LDSS + k]);
                nacc = fmaf(v.x, v.x, nacc);
                nacc = fmaf(v.y, v.y, nacc);
            }
        }
        __syncthreads();
    }

    if (wave < 3) {
        // C/D layout: VGPR v, lanes 0-15 -> M=v, lanes 16-31 -> M=v+8; N=lane%16
        #pragma unroll
        for (int v = 0; v < 8; ++v) {
            int row = tile * 16 + v + ((lane >> 4) << 3);
            int col = wave * 16 + m;
            dots[(size_t)row * NC + col] = acc[v];
        }
    } else {
        nacc += __shfl_xor(nacc, 16, 32);
        if (lane < 16) xnorm2[tile * 16 + lane] = nacc;
    }
}

// ---------------------------------------------------------------------------
// Kernel 2: brightness[i] = mean_r sqrt(||x_r||^2 - 2 dots[r][i] + ||p_i||^2)
// ---------------------------------------------------------------------------
__global__ void ff_bright(const float* __restrict__ dots, const float* __restrict__ xn,
                          const float* __restrict__ pn2, float* __restrict__ bri_ws,
                          float* __restrict__ bri_out) {
    __shared__ float red[256];
    int i = blockIdx.x, tid = threadIdx.x;
    const float p = pn2[i];
    float s = 0.f;
    for (int r = tid; r < ROWS; r += 256) {
        float d2 = xn[r] - 2.f * dots[(size_t)r * NC + i] + p;
        s += sqrtf(fmaxf(d2, 0.f));
    }
    red[tid] = s; __syncthreads();
    for (int o = 128; o > 0; o >>= 1) { if (tid < o) red[tid] += red[tid + o]; __syncthreads(); }
    if (tid == 0) {
        float b = red[0] * (1.f / (float)ROWS);
        bri_ws[i] = b;
        bri_out[i] = b;
    }
}

// ---------------------------------------------------------------------------
// Kernel 3: argmax + best update. Each thread redoes the tiny 35-way scan.
// ---------------------------------------------------------------------------
__global__ void ff_select(const float* __restrict__ bri, const float* __restrict__ pos,
                          const float* __restrict__ bestp_in, const float* __restrict__ bestb_in,
                          float* __restrict__ out_bp, float* __restrict__ ws_bp,
                          float* __restrict__ out_bb) {
    float bmax = bri[0]; int imax = 0;
    for (int i = 1; i < NF; ++i) { float b = bri[i]; if (b > bmax) { bmax = b; imax = i; } }
    bool take = bmax > bestb_in[0];
    int col = blockIdx.x * 256 + threadIdx.x;
    float v = take ? pos[(size_t)imax * HH + col] : bestp_in[col];
    out_bp[col] = v;
    ws_bp[col]  = v;   // 16B-aligned copy for the vectorized add kernel
    if (blockIdx.x == 0 && threadIdx.x == 0) out_bb[0] = take ? bmax : bestb_in[0];
}

// ---------------------------------------------------------------------------
// Kernel 4: output = x + new_best_position  (x should be L2-hot, 192MB L2)
// ---------------------------------------------------------------------------
__global__ __launch_bounds__(256)
void ff_add(const float* __restrict__ x, const float* __restrict__ bp,
            float* __restrict__ out) {
    size_t idx = ((size_t)blockIdx.x * 256 + threadIdx.x) * 4;
    int col = (int)(idx & (HH - 1));
    float4 xv = *reinterpret_cast<const float4*>(x + idx);
    float4 bv = *reinterpret_cast<const float4*>(bp + col);
    float4 o = make_float4(xv.x + bv.x, xv.y + bv.y, xv.z + bv.z, xv.w + bv.w);
    *reinterpret_cast<float4*>(out + idx) = o;
}

// ---------------------------------------------------------------------------
// Kernel 5: sequential Gauss-Seidel firefly update. Single 1024-thread block
// (32 waves). pi register-resident (4 f32/thread); d2 via shfl + one LDS hop.
// All branches are block-uniform (brightness in LDS), so barriers are safe.
// ---------------------------------------------------------------------------
__global__ __launch_bounds__(1024)
void ff_gauss(const float* __restrict__ pos_in, const float* __restrict__ noise,
              const float* __restrict__ bri, const float* __restrict__ base_p,
              const float* __restrict__ gamma_p, const float* __restrict__ alpha_p,
              float* __restrict__ outPos) {
    __shared__ float bl[NF];
    __shared__ float wred[32];
    const int tid = threadIdx.x, lane = tid & 31, wid = tid >> 5;

    for (int u = tid; u < NF * HH; u += 1024) outPos[u] = pos_in[u];
    if (tid < NF) bl[tid] = bri[tid];
    const float base  = base_p[0];
    const float gamma = gamma_p[0];
    const float alpha = alpha_p[0];
    __syncthreads();

    for (int i = 0; i < NF; ++i) {
        const float bi = bl[i];
        float pi[4];
        #pragma unroll
        for (int q = 0; q < 4; ++q) pi[q] = outPos[(size_t)i * HH + tid + q * 1024];

        for (int j = 0; j < NF; ++j) {
            if (j == i || !(bl[j] > bi)) continue;   // uniform scalar branch
            float pj[4]; float part = 0.f;
            #pragma unroll
            for (int q = 0; q < 4; ++q) {
                pj[q] = outPos[(size_t)j * HH + tid + q * 1024];
                float d = pi[q] - pj[q];
                part = fmaf(d, d, part);
            }
            #pragma unroll
            for (int o = 16; o > 0; o >>= 1) part += __shfl_xor(part, o, 32);
            if (lane == 0) wred[wid] = part;
            __syncthreads();
            if (wid == 0) {
                float r = wred[lane];
                #pragma unroll
                for (int o = 16; o > 0; o >>= 1) r += __shfl_xor(r, o, 32);
                if (lane == 0) wred[0] = r;
            }
            __syncthreads();
            const float d2 = wred[0];
            __syncthreads();
            const float attr = base * expf(-gamma * d2);
            #pragma unroll
            for (int q = 0; q < 4; ++q)
                pi[q] += attr * (pj[q] - pi[q]) +
                         alpha * noise[((size_t)i * NF + j) * HH + tid + q * 1024];
        }
        #pragma unroll
        for (int q = 0; q < 4; ++q) outPos[(size_t)i * HH + tid + q * 1024] = pi[q];
        __syncthreads();   // row i must be visible before later outer iters read it
    }
}

// ---------------------------------------------------------------------------
extern "C" void kernel_launch(void* const* d_in, const int* in_sizes, int n_in,
                              void* d_out, int out_size, void* d_ws, size_t ws_size,
                              hipStream_t stream) {
    (void)in_sizes; (void)n_in; (void)out_size; (void)ws_size;
    const float* x      = (const float*)d_in[0];
    const float* pos    = (const float*)d_in[1];
    const float* bestp  = (const float*)d_in[2];
    const float* bestb  = (const float*)d_in[3];
    const float* abase  = (const float*)d_in[4];
    const float* agamma = (const float*)d_in[5];
    const float* aalpha = (const float*)d_in[6];
    const float* noise  = (const float*)d_in[7];

    float* out      = (float*)d_out;
    float* out_x    = out;                       // 33,554,432
    float* out_pos  = out + 33554432;            // 143,360
    float* out_bri  = out + 33697792;            // 35
    float* out_bp   = out + 33697827;            // 4096 (not 16B aligned)
    float* out_bb   = out + 33701923;            // 1

    char*   ws    = (char*)d_ws;
    float2* bpack = (float2*)(ws);               // 786,432 B
    float*  dots  = (float*)(ws + 786432);       // 1,572,864 B
    float*  xn    = (float*)(ws + 2359296);      // 32,768 B
    float*  pn2   = (float*)(ws + 2392064);      // 140 B (pad 256)
    float*  briw  = (float*)(ws + 2392320);      // 140 B (pad 256)
    float*  bpw   = (float*)(ws + 2392576);      // 16,384 B (aligned best_position)

    ff_pack   <<<(NT * KSTEPS * 32 + 255) / 256, 256, 0, stream>>>(pos, bpack);
    ff_pnorm  <<<NF, 256, 0, stream>>>(pos, pn2);
    ff_gemm_norm<<<ROWS / 16, 128, 0, stream>>>(x, bpack, dots, xn);
    ff_bright <<<NF, 256, 0, stream>>>(dots, xn, pn2, briw, out_bri);
    ff_select <<<HH / 256, 256, 0, stream>>>(briw, pos, bestp, bestb, out_bp, bpw, out_bb);
    ff_add    <<<(ROWS * (HH / 4)) / 256, 256, 0, stream>>>(x, bpw, out_x);
    ff_gauss  <<<1, 1024, 0, stream>>>(pos, noise, briw, abase, agamma, aalpha, out_pos);
}